// TsallisBisectLoss_41077067219296
// MI455X (gfx1250) — compile-verified
//
#include <hip/hip_runtime.h>
#include <stdint.h>

#define ASL3 __attribute__((address_space(3)))
#define ASG1 __attribute__((address_space(1)))

typedef int v4i __attribute__((vector_size(16)));  // matches builtin's vector param

namespace {
constexpr int   kT     = 1024;   // 32 waves per block (wave32)
constexpr int   kVMax  = 32;     // register-resident elems/thread (d <= 32768)
constexpr float kAlpha = 1.5f;
constexpr int   kIter  = 15;
}

__device__ __forceinline__ float warp_sum32(float v) {
  v += __shfl_xor(v, 16, 32);
  v += __shfl_xor(v, 8, 32);
  v += __shfl_xor(v, 4, 32);
  v += __shfl_xor(v, 2, 32);
  v += __shfl_xor(v, 1, 32);
  return v;
}

__device__ __forceinline__ float warp_max32(float v) {
  v = fmaxf(v, __shfl_xor(v, 16, 32));
  v = fmaxf(v, __shfl_xor(v, 8, 32));
  v = fmaxf(v, __shfl_xor(v, 4, 32));
  v = fmaxf(v, __shfl_xor(v, 2, 32));
  v = fmaxf(v, __shfl_xor(v, 1, 32));
  return v;
}

// Block-wide reduce (1024 threads = 32 waves); result broadcast to all threads.
__device__ __forceinline__ float block_sum(float v, float* red) {
  v = warp_sum32(v);
  const int lane = threadIdx.x & 31;
  const int wave = threadIdx.x >> 5;
  if (lane == 0) red[wave] = v;
  __syncthreads();
  float r = red[lane];          // 32 waves -> red[0..31] all valid
  r = warp_sum32(r);            // every wave redundantly reduces -> broadcast
  __syncthreads();              // protect red[] for next use
  return r;
}

__device__ __forceinline__ float block_max(float v, float* red) {
  v = warp_max32(v);
  const int lane = threadIdx.x & 31;
  const int wave = threadIdx.x >> 5;
  if (lane == 0) red[wave] = v;
  __syncthreads();
  float r = red[lane];
  r = warp_max32(r);
  __syncthreads();
  return r;
}

__global__ __launch_bounds__(kT)
void tsallis_row_kernel(const float* __restrict__ X,
                        const int*   __restrict__ target,
                        float*       __restrict__ row_loss,
                        int n, int d) {
  extern __shared__ float srow[];   // d floats: one full row of raw logits
  __shared__ float red[32];

  const int tid = threadIdx.x;
  const int row = blockIdx.x;
  const float* g = X + (size_t)row * (size_t)d;

  // ---- Stage the row into LDS with the CDNA5 async copy path (B128) ----
  const int nchunk = d >> 2;  // 16-byte chunks
  for (int c = tid; c < nchunk; c += kT) {
#if __has_builtin(__builtin_amdgcn_global_load_async_to_lds_b128)
    __builtin_amdgcn_global_load_async_to_lds_b128(
        (ASG1 v4i*)(uintptr_t)(g + 4 * c),
        (ASL3 v4i*)(uintptr_t)(srow + 4 * c),
        /*offset=*/0, /*cpol=*/0);
#else
    uint32_t lds_off = (uint32_t)(uintptr_t)(srow + 4 * c);
    asm volatile("global_load_async_to_lds_b128 %0, %1, %2"
                 :: "v"(lds_off), "v"(c * 16), "s"(g) : "memory");
#endif
  }
  // scalar tail if d % 4 != 0 (not hit for d = 32000)
  for (int e = (nchunk << 2) + tid; e < d; e += kT) srow[e] = g[e];
#if __has_builtin(__builtin_amdgcn_s_wait_asynccnt)
  __builtin_amdgcn_s_wait_asynccnt(0);
#else
  asm volatile("s_wait_asynccnt 0" ::: "memory");
#endif
  __syncthreads();  // all waves' async writes now visible

  // ---- Register-resident scaled logits: x = (alpha-1) * X ----
  float x[kVMax];
  float mx = -3.0e38f;
#pragma unroll
  for (int j = 0; j < kVMax; ++j) {
    const int e = j * kT + tid;
    x[j] = (e < d) ? 0.5f * srow[e] : -1.0e30f;  // finite sentinel: relu()=0 always
    mx = fmaxf(mx, x[j]);
  }
  mx = block_max(mx, red);

  float tau_lo = mx - 1.0f;                       // gp(1) = 1
  float dm = 1.0f - 1.0f / sqrtf((float)d);       // tau_hi - tau_lo, (1/d)^(a-1)

  // f_lo = sum relu(x - tau_lo)^2 - 1
  float acc = 0.0f;
#pragma unroll
  for (int j = 0; j < kVMax; ++j) {
    const float t = fmaxf(x[j] - tau_lo, 0.0f);
    acc = fmaf(t, t, acc);
  }
  const float f_lo = block_sum(acc, red) - 1.0f;

  // ---- 15 bisection steps, all out of registers ----
  float tau_m = tau_lo;
  for (int it = 0; it < kIter; ++it) {
    dm *= 0.5f;
    tau_m = tau_lo + dm;
    float a = 0.0f;
#pragma unroll
    for (int j = 0; j < kVMax; ++j) {
      const float t = fmaxf(x[j] - tau_m, 0.0f);
      a = fmaf(t, t, a);
    }
    const float f_m = block_sum(a, red) - 1.0f;   // uniform across block
    if (f_m * f_lo >= 0.0f) tau_lo = tau_m;       // uniform branch
  }

  // ---- Final pass with last tau_m: p = t^2, p^1.5 = t^3, dot(p, X) ----
  float s3 = 0.0f, sdot = 0.0f;
#pragma unroll
  for (int j = 0; j < kVMax; ++j) {
    const float t = fmaxf(x[j] - tau_m, 0.0f);
    const float p = t * t;
    s3   = fmaf(p, t, s3);       // sum p^alpha
    sdot = fmaf(p, x[j], sdot);  // sum p * (0.5 X)
  }
  s3   = block_sum(s3, red);
  sdot = block_sum(sdot, red);

  if (tid == 0) {
    const int tgt = target[row];
    const float xt = srow[tgt];  // original (unscaled) X[row, tgt]
    const float loss =
        (1.0f - s3) * (1.0f / (kAlpha * (kAlpha - 1.0f)))  // /0.75
        + 2.0f * sdot                                      // p . X (X = 2*x)
        - xt;                                              // one-hot subtraction
    row_loss[row] = loss;
  }
}

__global__ __launch_bounds__(kT)
void reduce_mean_kernel(const float* __restrict__ row_loss,
                        float* __restrict__ out, int n) {
  __shared__ float red[32];
  float v = 0.0f;
  for (int i = threadIdx.x; i < n; i += kT) v += row_loss[i];
  v = block_sum(v, red);
  if (threadIdx.x == 0) out[0] = v / (float)n;
}

extern "C" void kernel_launch(void* const* d_in, const int* in_sizes, int n_in,
                              void* d_out, int out_size, void* d_ws, size_t ws_size,
                              hipStream_t stream) {
  const float* X   = (const float*)d_in[0];
  const int*   tgt = (const int*)d_in[1];
  const int n = in_sizes[1];                 // 4096 rows
  const int d = in_sizes[0] / n;             // 32000 logits/row
  float* row_loss = (float*)d_ws;            // n floats of scratch

  const size_t shbytes = (size_t)d * sizeof(float);  // 128 KB < 320 KB WGP LDS
  tsallis_row_kernel<<<n, kT, shbytes, stream>>>(X, tgt, row_loss, n, d);
  reduce_mean_kernel<<<1, kT, 0, stream>>>(row_loss, (float*)d_out, n);
}